// Stage3CKALoss_17944373363084
// MI455X (gfx1250) — compile-verified
//
#include <hip/hip_runtime.h>
#include <hip/hip_bf16.h>

#define B_   16
#define CS_  512
#define CT_  768
#define H_   96
#define W_   96
#define WS_  16
#define K_   10
#define WWIN 81
#define NWIN (WWIN*WWIN)   /* 6561 */
#define NTOK 256
#define HW_  (H_*W_)
#define EPS_ 1e-6f
#define NEGINF (-3.0e38f)

#define CHUNK 32
#define LSTR  34           /* padded LDS row stride (floats), bank-conflict free */
#define IROWS 4            /* gram tile rows per workgroup */
#define IBLK  (16/IROWS)   /* workgroups per (b,k) */
#define AUXSZ 514          /* per-(b,k) aux floats: u_s[256], u_t[256], ss_s, ss_t */

typedef float v2f __attribute__((ext_vector_type(2)));
typedef float v8f __attribute__((ext_vector_type(8)));

// ---------------------------------------------------------------------------
// Pass 1: per-pixel channel reductions: ||f||^2, f·f_right, f·f_down  (s and t)
// ---------------------------------------------------------------------------
__global__ __launch_bounds__(256) void k_dots(const float* __restrict__ s,
                                              const float* __restrict__ t,
                                              float* __restrict__ dots) {
  int p = blockIdx.x * 256 + threadIdx.x;
  const int N = B_ * HW_;
  if (p >= N) return;
  int b  = p / HW_;
  int hw = p - b * HW_;
  int h  = hw / W_;
  int w  = hw - h * W_;
  const bool rok = (w < W_ - 1);
  const bool dok = (h < H_ - 1);

  float ns = 0.f, rs = 0.f, dsv = 0.f;
  {
    const float* f = s + (size_t)b * CS_ * HW_ + hw;
    for (int c = 0; c < CS_; ++c, f += HW_) {
      float v = *f;
      ns += v * v;
      if (rok) rs  += v * f[1];
      if (dok) dsv += v * f[W_];
    }
  }
  float nt = 0.f, rt = 0.f, dt = 0.f;
  {
    const float* f = t + (size_t)b * CT_ * HW_ + hw;
    for (int c = 0; c < CT_; ++c, f += HW_) {
      float v = *f;
      nt += v * v;
      if (rok) rt += v * f[1];
      if (dok) dt += v * f[W_];
    }
  }
  dots[0*N + p] = ns;  dots[1*N + p] = rs;  dots[2*N + p] = dsv;
  dots[3*N + p] = nt;  dots[4*N + p] = rt;  dots[5*N + p] = dt;
}

// ---------------------------------------------------------------------------
// Pass 2: divergence map
// ---------------------------------------------------------------------------
__global__ __launch_bounds__(256) void k_div(const float* __restrict__ dots,
                                             float* __restrict__ divm) {
  int p = blockIdx.x * 256 + threadIdx.x;
  const int N = B_ * HW_;
  if (p >= N) return;
  int hw = p % HW_;
  int h  = hw / W_;
  int w  = hw - h * W_;

  float n0 = fmaxf(sqrtf(dots[0*N + p]), EPS_);
  float m0 = fmaxf(sqrtf(dots[3*N + p]), EPS_);
  float d = 0.f;
  if (w < W_ - 1) {
    float n1 = fmaxf(sqrtf(dots[0*N + p + 1]), EPS_);
    float m1 = fmaxf(sqrtf(dots[3*N + p + 1]), EPS_);
    float cs = dots[1*N + p] / (n0 * n1);
    float ct = dots[4*N + p] / (m0 * m1);
    d += fabsf(ct - cs);
  }
  if (h < H_ - 1) {
    float n1 = fmaxf(sqrtf(dots[0*N + p + W_]), EPS_);
    float m1 = fmaxf(sqrtf(dots[3*N + p + W_]), EPS_);
    float cs = dots[2*N + p] / (n0 * n1);
    float ct = dots[5*N + p] / (m0 * m1);
    d += fabsf(ct - cs);
  }
  divm[p] = d;
}

// ---------------------------------------------------------------------------
// Pass 3: separable 16x16 sum pooling (VALID)
// ---------------------------------------------------------------------------
__global__ __launch_bounds__(256) void k_rowsum(const float* __restrict__ divm,
                                                float* __restrict__ rowsum) {
  int p = blockIdx.x * 256 + threadIdx.x;
  const int total = B_ * H_ * WWIN;
  if (p >= total) return;
  int b  = p / (H_ * WWIN);
  int r  = p - b * (H_ * WWIN);
  int h  = r / WWIN;
  int wo = r - h * WWIN;
  const float* src = divm + b * HW_ + h * W_ + wo;
  float s = 0.f;
#pragma unroll
  for (int i = 0; i < WS_; ++i) s += src[i];
  rowsum[p] = s;
}

__global__ __launch_bounds__(256) void k_colsum(const float* __restrict__ rowsum,
                                               float* __restrict__ score) {
  int p = blockIdx.x * 256 + threadIdx.x;
  const int total = B_ * NWIN;
  if (p >= total) return;
  int b  = p / NWIN;
  int r  = p - b * NWIN;
  int ho = r / WWIN;
  int wo = r - ho * WWIN;
  const float* src = rowsum + (b * H_ + ho) * WWIN + wo;
  float s = 0.f;
#pragma unroll
  for (int i = 0; i < WS_; ++i) s += src[i * WWIN];
  score[p] = s;
}

// ---------------------------------------------------------------------------
// Pass 4: greedy NMS (iterated argmax + suppression), stable tie-break.
// ---------------------------------------------------------------------------
__global__ __launch_bounds__(256) void k_nms(const float* __restrict__ score,
                                             int* __restrict__ sidx,
                                             float* __restrict__ wscore) {
  __shared__ float sc[NWIN];
  __shared__ float rbest[256];
  __shared__ int   ridx[256];
  const int b = blockIdx.x, tid = threadIdx.x;
  const float* s = score + b * NWIN;
  for (int p = tid; p < NWIN; p += 256) sc[p] = s[p];
  __syncthreads();

  for (int sel = 0; sel < K_; ++sel) {
    float best = NEGINF; int bidx = NWIN;
    for (int p = tid; p < NWIN; p += 256) {
      float v = sc[p];
      if (v > best || (v == best && p < bidx)) { best = v; bidx = p; }
    }
    rbest[tid] = best; ridx[tid] = bidx;
    __syncthreads();
    for (int off = 128; off; off >>= 1) {
      if (tid < off) {
        float v = rbest[tid + off]; int ii = ridx[tid + off];
        if (v > rbest[tid] || (v == rbest[tid] && ii < ridx[tid])) {
          rbest[tid] = v; ridx[tid] = ii;
        }
      }
      __syncthreads();
    }
    const int ch = ridx[0];
    if (tid == 0) { sidx[b * K_ + sel] = ch; wscore[b * K_ + sel] = rbest[0]; }
    const int cy = ch / WWIN, cx = ch % WWIN;
    __syncthreads();
    for (int p = tid; p < NWIN; p += 256) {
      int y = p / WWIN, x = p - y * WWIN;
      int dy = y > cy ? y - cy : cy - y;
      int dx = x > cx ? x - cx : cx - x;
      int iy = WS_ - dy > 0 ? WS_ - dy : 0;
      int ix = WS_ - dx > 0 ? WS_ - dx : 0;
      float inter = (float)(iy * ix);
      float iou = inter / fmaxf(512.f - inter, EPS_);
      if (iou > 0.2f) sc[p] = NEGINF;
    }
    __syncthreads();
  }
}

// ---------------------------------------------------------------------------
// Pass 5: per-(b,k) centering aux:  mu (LDS only), u = X*mu per token, mu'mu.
// aux layout per bk: [u_s(256) | u_t(256) | ss_s | ss_t]
// ---------------------------------------------------------------------------
__global__ __launch_bounds__(256) void k_aux(const float* __restrict__ sfeat,
                                             const float* __restrict__ tfeat,
                                             const int* __restrict__ sidx,
                                             float* __restrict__ aux) {
  __shared__ float mu[CS_ + CT_];
  __shared__ float red[256];
  const int bk = blockIdx.x;
  const int b  = bk / K_;
  const int tid = threadIdx.x;
  const int widx = sidx[bk];
  const int y0 = widx / WWIN, x0 = widx % WWIN;

  for (int c = tid; c < CS_ + CT_; c += 256) {
    const float* f = (c < CS_)
        ? sfeat + ((size_t)b * CS_ + c) * HW_
        : tfeat + ((size_t)b * CT_ + (c - CS_)) * HW_;
    const float* w0 = f + y0 * W_ + x0;
    float sum = 0.f;
#pragma unroll
    for (int dy = 0; dy < WS_; ++dy)
#pragma unroll
      for (int dx = 0; dx < WS_; ++dx) sum += w0[dy * W_ + dx];
    mu[c] = sum * (1.0f / 256.0f);
  }
  __syncthreads();

  // u[n] = sum_c X[n,c]*mu[c]   (one token per thread; coalesced over lanes)
  const int ty = tid >> 4, tx = tid & 15;
  {
    const float* p = sfeat + (size_t)b * CS_ * HW_ + (y0 + ty) * W_ + (x0 + tx);
    float u = 0.f;
    for (int c = 0; c < CS_; ++c, p += HW_) u += (*p) * mu[c];
    aux[(size_t)bk * AUXSZ + tid] = u;
  }
  {
    const float* p = tfeat + (size_t)b * CT_ * HW_ + (y0 + ty) * W_ + (x0 + tx);
    float u = 0.f;
    for (int c = 0; c < CT_; ++c, p += HW_) u += (*p) * mu[CS_ + c];
    aux[(size_t)bk * AUXSZ + 256 + tid] = u;
  }

  // ss = mu'mu for each feature
  float v = 0.f;
  for (int c = tid; c < CS_; c += 256) v += mu[c] * mu[c];
  red[tid] = v;
  __syncthreads();
  for (int off = 128; off; off >>= 1) {
    if (tid < off) red[tid] += red[tid + off];
    __syncthreads();
  }
  if (tid == 0) aux[(size_t)bk * AUXSZ + 512] = red[0];
  __syncthreads();
  v = 0.f;
  for (int c = CS_ + tid; c < CS_ + CT_; c += 256) v += mu[c] * mu[c];
  red[tid] = v;
  __syncthreads();
  for (int off = 128; off; off >>= 1) {
    if (tid < off) red[tid] += red[tid + off];
    __syncthreads();
  }
  if (tid == 0) aux[(size_t)bk * AUXSZ + 513] = red[0];
}

// ---------------------------------------------------------------------------
// Pass 6: raw gram + fused HSIC via fp32 WMMA.
// Block = (bk, ih). 16 waves; wave w owns tiles (ibase..ibase+3, j=w).
// Chunks of 32 channels staged in LDS (pure copy; centering applied
// algebraically at the end:  Gc = G0 - u[n] - u[m] + mu'mu).
// ---------------------------------------------------------------------------
__device__ __forceinline__ void accum_gram4(const float* __restrict__ fb,
                                            int Cdim, int y0, int x0,
                                            float* chunkbuf,
                                            int ibase, int wv, int m, int kh,
                                            int tid, v8f g[IROWS]) {
  const int tok = tid & 255;
  const int c0  = tid >> 8;              // 0 or 1
  const int ty  = tok >> 4, tx = tok & 15;
  const float* gbase = fb + (size_t)c0 * HW_ + (y0 + ty) * W_ + (x0 + tx);
  float* lbase = chunkbuf + tok * LSTR + c0;

  const float* brow  = &chunkbuf[(wv * 16 + m) * LSTR + kh];
  const float* arow0 = &chunkbuf[(ibase * 16 + m) * LSTR + kh];

  for (int kc = 0; kc < Cdim; kc += CHUNK) {
    __syncthreads();
    {
      const float* gp = gbase + (size_t)kc * HW_;
#pragma unroll
      for (int q = 0; q < CHUNK / 2; ++q) {
        lbase[2 * q] = *gp;            // pure copy, pointer-march addressing
        gp += 2 * HW_;
      }
    }
    __syncthreads();
#pragma unroll
    for (int kk = 0; kk < CHUNK; kk += 4) {
      v2f bb;
      bb.x = brow[kk]; bb.y = brow[kk + 1];
#pragma unroll
      for (int ii = 0; ii < IROWS; ++ii) {
        v2f a;
        a.x = arow0[ii * 16 * LSTR + kk];
        a.y = arow0[ii * 16 * LSTR + kk + 1];
        g[ii] = __builtin_amdgcn_wmma_f32_16x16x4_f32(false, a, false, bb,
                                                      (short)0, g[ii],
                                                      false, false);
      }
    }
  }
}

__global__ __launch_bounds__(512) void k_gram(const float* __restrict__ sfeat,
                                              const float* __restrict__ tfeat,
                                              const int* __restrict__ sidx,
                                              const float* __restrict__ aux,
                                              float* __restrict__ part) {
  __shared__ float chunkbuf[NTOK * LSTR];
  __shared__ float wred[16 * 3];

  const int blk = blockIdx.x;
  const int ih  = blk & (IBLK - 1);
  const int bk  = blk / IBLK;
  const int b   = bk / K_;

  const int widx = sidx[bk];
  const int y0 = widx / WWIN, x0 = widx % WWIN;

  const int tid  = threadIdx.x;
  const int lane = tid & 31;
  const int wv   = tid >> 5;               // wave id == tile col j
  const int m    = lane & 15;              // row within 16x16 tile
  const int kh   = (lane >> 4) << 1;       // K sub-offset 0 or 2
  const int ibase = ih * IROWS;

  v8f gx[IROWS], gy[IROWS];
#pragma unroll
  for (int ii = 0; ii < IROWS; ++ii) {
    gx[ii] = (v8f){0.f,0.f,0.f,0.f,0.f,0.f,0.f,0.f};
    gy[ii] = (v8f){0.f,0.f,0.f,0.f,0.f,0.f,0.f,0.f};
  }

  accum_gram4(sfeat + (size_t)b * CS_ * HW_, CS_, y0, x0,
              chunkbuf, ibase, wv, m, kh, tid, gx);
  accum_gram4(tfeat + (size_t)b * CT_ * HW_, CT_, y0, x0,
              chunkbuf, ibase, wv, m, kh, tid, gy);

  // centering correction + fused HSIC / ||gx||^2 / ||gy||^2
  const float* auxp = aux + (size_t)bk * AUXSZ;
  const float* us = auxp;
  const float* ut = auxp + 256;
  const float ss_s = auxp[512];
  const float ss_t = auxp[513];
  const int nL = lane & 15;
  const float usj = us[wv * 16 + nL];
  const float utj = ut[wv * 16 + nL];
  const int mhalf = (lane >> 4) << 3;      // 0 or 8 (C/D layout row half)

  float hs = 0.f, nx = 0.f, ny = 0.f;
#pragma unroll
  for (int ii = 0; ii < IROWS; ++ii) {
    const int rowb = (ibase + ii) * 16 + mhalf;
#pragma unroll
    for (int r = 0; r < 8; ++r) {
      float gxv = gx[ii][r] - us[rowb + r] - usj + ss_s;
      float gyv = gy[ii][r] - ut[rowb + r] - utj + ss_t;
      hs += gxv * gyv;
      nx += gxv * gxv;
      ny += gyv * gyv;
    }
  }
#pragma unroll
  for (int off = 16; off > 0; off >>= 1) {
    hs += __shfl_down(hs, off);
    nx += __shfl_down(nx, off);
    ny += __shfl_down(ny, off);
  }
  __syncthreads();
  if (lane == 0) { wred[wv*3+0] = hs; wred[wv*3+1] = nx; wred[wv*3+2] = ny; }
  __syncthreads();
  if (tid == 0) {
    float a = 0.f, c1 = 0.f, c2 = 0.f;
    for (int q = 0; q < 16; ++q) { a += wred[q*3]; c1 += wred[q*3+1]; c2 += wred[q*3+2]; }
    part[blk*3+0] = a; part[blk*3+1] = c1; part[blk*3+2] = c2;
  }
}

// ---------------------------------------------------------------------------
// Pass 7: finalize CKA + loss.  out = [loss, cka(160), win_scores(160)]
// ---------------------------------------------------------------------------
__global__ __launch_bounds__(256) void k_final(const float* __restrict__ part,
                                               float* __restrict__ out) {
  __shared__ float red[256];
  const int tid = threadIdx.x;
  float contrib = 0.f;
  if (tid < B_ * K_) {
    float hs = 0.f, nx2 = 0.f, ny2 = 0.f;
    for (int i = 0; i < IBLK; ++i) {
      const float* p = part + (size_t)(tid * IBLK + i) * 3;
      hs += p[0]; nx2 += p[1]; ny2 += p[2];
    }
    float nx = sqrtf(fmaxf(nx2, EPS_));
    float ny = sqrtf(fmaxf(ny2, EPS_));
    float cka = hs / fmaxf(nx * ny, EPS_);
    cka = fminf(1.f, fmaxf(-1.f, cka));
    out[1 + tid] = cka;
    contrib = 1.f - cka;
  }
  red[tid] = contrib;
  __syncthreads();
  for (int off = 128; off; off >>= 1) {
    if (tid < off) red[tid] += red[tid + off];
    __syncthreads();
  }
  if (tid == 0) out[0] = red[0] / (float)(B_ * K_);
}

// ---------------------------------------------------------------------------
extern "C" void kernel_launch(void* const* d_in, const int* in_sizes, int n_in,
                              void* d_out, int out_size, void* d_ws, size_t ws_size,
                              hipStream_t stream) {
  (void)in_sizes; (void)n_in; (void)out_size; (void)ws_size;
  const float* sfeat = (const float*)d_in[0];
  const float* tfeat = (const float*)d_in[1];
  float* out = (float*)d_out;
  float* ws  = (float*)d_ws;

  const int N = B_ * HW_;
  float* dots   = ws;                                  // 6*N
  float* divm   = dots + (size_t)6 * N;                // N
  float* rowsum = divm + N;                            // B*96*81
  float* score  = rowsum + B_ * H_ * WWIN;             // B*81*81
  int*   sidx   = (int*)(score + B_ * NWIN);           // B*K ints
  float* aux    = (float*)(sidx + B_ * K_);            // B*K*AUXSZ
  float* part   = aux + (size_t)B_ * K_ * AUXSZ;       // B*K*IBLK*3

  k_dots  <<<(N + 255) / 256, 256, 0, stream>>>(sfeat, tfeat, dots);
  k_div   <<<(N + 255) / 256, 256, 0, stream>>>(dots, divm);
  k_rowsum<<<(B_ * H_ * WWIN + 255) / 256, 256, 0, stream>>>(divm, rowsum);
  k_colsum<<<(B_ * NWIN + 255) / 256, 256, 0, stream>>>(rowsum, score);
  k_nms   <<<B_, 256, 0, stream>>>(score, sidx, out + 1 + B_ * K_);
  k_aux   <<<B_ * K_, 256, 0, stream>>>(sfeat, tfeat, sidx, aux);
  k_gram  <<<B_ * K_ * IBLK, 512, 0, stream>>>(sfeat, tfeat, sidx, aux, part);
  k_final <<<1, 256, 0, stream>>>(part, out);
}